// Tree_SSM_86139864088936
// MI455X (gfx1250) — compile-verified
//
#include <hip/hip_runtime.h>
#include <hip/hip_bf16.h>
#include <math.h>

#define BSZ 8
#define HH  64
#define WWID 64
#define NL  4096            // H*W
#define DM  96              // d_model
#define DI  192             // d_inner
#define DTR 6
#define NE  8064            // grid edges: 64*63 + 63*64
#define NEP 8192            // padded to pow2 for bitonic sort
#define NTOK (BSZ*NL)       // 32768 tokens

typedef __attribute__((ext_vector_type(2))) float v2f;
typedef __attribute__((ext_vector_type(8))) float v8f;

__device__ __forceinline__ float silu_f(float v) { return v / (1.0f + expf(-v)); }
__device__ __forceinline__ float softplus_f(float v) {
  return fmaxf(v, 0.0f) + log1pf(expf(-fabsf(v)));
}

// ---------------------------------------------------------------------------
// 1) in_proj GEMM: X[32768,96] @ W^T (W=(384,96)) -> xz[32768,384], fp32 WMMA
// ---------------------------------------------------------------------------
__global__ void in_proj_gemm_kernel(const float* __restrict__ X,
                                    const float* __restrict__ Wp,   // (384,96)
                                    float* __restrict__ xz) {
  const int NT = (2 * DI) / 16;                       // 24 N-tiles
  int wave = blockIdx.x * (blockDim.x >> 5) + (threadIdx.x >> 5);
  int lane = threadIdx.x & 31;
  int tm = (wave / NT) * 16;
  int tn = (wave % NT) * 16;
  int row  = lane & 15;     // M row for A, N col for B
  int half = lane >> 4;
  v8f acc = {0.f,0.f,0.f,0.f,0.f,0.f,0.f,0.f};
  const float* arow = X  + (size_t)(tm + row) * DM;
  const float* brow = Wp + (size_t)(tn + row) * DM;   // B[k][n] = Wp[n][k]
  for (int k = 0; k < DM; k += 4) {
    int ka = k + half * 2;
    v2f a;  a.x  = arow[ka]; a.y  = arow[ka + 1];
    v2f bv; bv.x = brow[ka]; bv.y = brow[ka + 1];
    acc = __builtin_amdgcn_wmma_f32_16x16x4_f32(false, a, false, bv,
                                                (short)0, acc, false, false);
  }
  for (int g = 0; g < 8; ++g) {
    int r = tm + g + half * 8;
    xz[(size_t)r * (2 * DI) + tn + row] = acc[g];
  }
}

// ---------------------------------------------------------------------------
// 2) z = silu(xz[:,192:384]) stored (b,l,c)
// ---------------------------------------------------------------------------
__global__ void z_silu_kernel(const float* __restrict__ xz, float* __restrict__ zs) {
  int gid = blockIdx.x * blockDim.x + threadIdx.x;
  if (gid >= NTOK * DI) return;
  int bl = gid / DI, c = gid % DI;
  float v = xz[(size_t)bl * (2 * DI) + DI + c];
  zs[gid] = silu_f(v);
}

// ---------------------------------------------------------------------------
// 3) depthwise 3x3 conv (SAME) + bias + silu  -> xc laid out (b,d,l)
// ---------------------------------------------------------------------------
__global__ void dwconv_silu_kernel(const float* __restrict__ xz,
                                   const float* __restrict__ cw,  // (192,1,3,3)
                                   const float* __restrict__ cb,
                                   float* __restrict__ xc) {
  int gid = blockIdx.x * blockDim.x + threadIdx.x;
  if (gid >= BSZ * DI * NL) return;
  int l = gid % NL;
  int d = (gid / NL) % DI;
  int b = gid / (NL * DI);
  int h = l / WWID, w = l % WWID;
  float acc = cb[d];
  for (int i = 0; i < 3; ++i) {
    int hh = h + i - 1;
    if (hh < 0 || hh >= HH) continue;
    for (int j = 0; j < 3; ++j) {
      int ww = w + j - 1;
      if (ww < 0 || ww >= WWID) continue;
      float xin = xz[((size_t)b * NL + hh * WWID + ww) * (2 * DI) + d];
      acc += cw[d * 9 + i * 3 + j] * xin;
    }
  }
  xc[((size_t)b * DI + d) * NL + l] = silu_f(acc);
}

// ---------------------------------------------------------------------------
// 4) per-node inverse L2 norm over channels (fp64, matches reference)
// ---------------------------------------------------------------------------
__global__ void node_norm_kernel(const float* __restrict__ xc, double* __restrict__ invn) {
  int gid = blockIdx.x * blockDim.x + threadIdx.x;
  if (gid >= BSZ * NL) return;
  int b = gid / NL, v = gid % NL;
  double s = 0.0;
  for (int d = 0; d < DI; ++d) {
    double f = (double)xc[((size_t)b * DI + d) * NL + v];
    s += f * f;
  }
  double nrm = sqrt(s);
  invn[gid] = 1.0 / fmax(nrm, 1e-8);
}

__device__ __forceinline__ void edge_uv(int e, int& u, int& v) {
  if (e < HH * (WWID - 1)) {                 // horizontal
    int r = e / (WWID - 1), c = e % (WWID - 1);
    u = r * WWID + c; v = u + 1;
  } else {                                   // vertical
    int e2 = e - HH * (WWID - 1);
    int r = e2 / WWID, c = e2 % WWID;
    u = r * WWID + c; v = u + WWID;
  }
}

// 5) edge weights w = exp(-dot(fn_u, fn_v)), fp64; pad keys to 8192
__global__ void edge_weight_kernel(const float* __restrict__ xc,
                                   const double* __restrict__ invn,
                                   double* __restrict__ keys, int* __restrict__ eidx) {
  int gid = blockIdx.x * blockDim.x + threadIdx.x;
  if (gid >= BSZ * NEP) return;
  int b = gid / NEP, e = gid % NEP;
  if (e >= NE) { keys[gid] = 1e300; eidx[gid] = e; return; }
  int u, v; edge_uv(e, u, v);
  double dot = 0.0;
  const float* base = xc + (size_t)b * DI * NL;
  for (int d = 0; d < DI; ++d)
    dot += (double)base[(size_t)d * NL + u] * (double)base[(size_t)d * NL + v];
  dot *= invn[b * NL + u] * invn[b * NL + v];
  keys[gid] = exp(-dot);
  eidx[gid] = e;
}

// 6) stable ascending sort (bitonic, tie-break by edge index), 1 block/batch
__global__ void sort_edges_kernel(double* __restrict__ keys, int* __restrict__ eidx) {
  int b = blockIdx.x;
  double* k = keys + (size_t)b * NEP;
  int* id = eidx + (size_t)b * NEP;
  for (int len = 2; len <= NEP; len <<= 1) {
    for (int j = len >> 1; j > 0; j >>= 1) {
      for (int i = threadIdx.x; i < NEP; i += blockDim.x) {
        int ixj = i ^ j;
        if (ixj > i) {
          bool asc = ((i & len) == 0);
          double a = k[i], c = k[ixj];
          int ia = id[i], ic = id[ixj];
          bool gt = (a > c) || (a == c && ia > ic);
          if (gt == asc) { k[i] = c; k[ixj] = a; id[i] = ic; id[ixj] = ia; }
        }
      }
      __syncthreads();
    }
  }
}

// ---------------------------------------------------------------------------
// 7) Kruskal MST + BFS (serial per batch; BFS order is depth-sorted)
// ---------------------------------------------------------------------------
__global__ void build_tree_kernel(const int* __restrict__ eidx,
                                  int* __restrict__ parent, int* __restrict__ dep,
                                  int* __restrict__ order, int* __restrict__ lvls,
                                  int* __restrict__ uf, int* __restrict__ adjh,
                                  int* __restrict__ adjn, int* __restrict__ adjt,
                                  int* __restrict__ mdep) {
  int b = blockIdx.x * blockDim.x + threadIdx.x;
  if (b >= BSZ) return;
  const int n = NL;
  const int* id = eidx + (size_t)b * NEP;
  int* UF  = uf     + (size_t)b * n;
  int* HD  = adjh   + (size_t)b * n;
  int* NX  = adjn   + (size_t)b * NEP;
  int* TO  = adjt   + (size_t)b * NEP;
  int* PAR = parent + (size_t)b * n;
  int* DEP = dep    + (size_t)b * n;
  int* ORD = order  + (size_t)b * n;
  int* LS  = lvls   + (size_t)b * 4100;
  for (int i = 0; i < n; ++i) { UF[i] = i; HD[i] = -1; DEP[i] = -1; PAR[i] = 0; }
  auto find = [&](int a) {
    while (UF[a] != a) { UF[a] = UF[UF[a]]; a = UF[a]; }
    return a;
  };
  int cnt = 0, slot = 0;
  for (int t = 0; t < NEP && cnt < n - 1; ++t) {
    int e = id[t];
    if (e >= NE) continue;
    int u, v; edge_uv(e, u, v);
    int ru = find(u), rv = find(v);
    if (ru != rv) {
      UF[ru] = rv;
      TO[slot] = v; NX[slot] = HD[u]; HD[u] = slot; ++slot;
      TO[slot] = u; NX[slot] = HD[v]; HD[v] = slot; ++slot;
      ++cnt;
    }
  }
  // BFS from node 0
  DEP[0] = 0; ORD[0] = 0;
  int head = 0, tail = 1;
  while (head < tail) {
    int a = ORD[head++];
    for (int s = HD[a]; s != -1; s = NX[s]) {
      int nb = TO[s];
      if (DEP[nb] < 0) { DEP[nb] = DEP[a] + 1; PAR[nb] = a; ORD[tail++] = nb; }
    }
  }
  int md = 0;
  for (int i = 0; i < n; ++i) {
    int d = DEP[ORD[i]];
    if (i == 0 || d != DEP[ORD[i - 1]]) LS[d] = i;
    if (d > md) md = d;
  }
  LS[md + 1] = n;
  mdep[b] = md;
}

// ---------------------------------------------------------------------------
// 8) x_dbl / dts / softplus / deltaA(ew) / BX — one thread per (b,l)
// ---------------------------------------------------------------------------
__global__ void ssm_pre_kernel(const float* __restrict__ xc,
                               const float* __restrict__ xpw,   // (8,192)
                               const float* __restrict__ dtw,   // (192,6)
                               const float* __restrict__ dtb,   // (192)
                               const float* __restrict__ alog,  // (192,1)
                               float* __restrict__ ew, float* __restrict__ bx,
                               float* __restrict__ cs) {
  int gid = blockIdx.x * blockDim.x + threadIdx.x;
  if (gid >= NTOK) return;
  int b = gid / NL, l = gid % NL;
  const float* xs = xc + (size_t)b * DI * NL;
  float xdbl[DTR + 2];
  for (int c = 0; c < DTR + 2; ++c) {
    float s = 0.f;
    for (int d = 0; d < DI; ++d) s += xpw[c * DI + d] * xs[(size_t)d * NL + l];
    xdbl[c] = s;
  }
  float Bv = xdbl[DTR], Cv = xdbl[DTR + 1];
  cs[gid] = Cv;
  for (int d = 0; d < DI; ++d) {
    float dt = dtb[d];
    for (int r = 0; r < DTR; ++r) dt += dtw[d * DTR + r] * xdbl[r];
    dt = softplus_f(dt);
    float Aval = -expf(alog[d]);
    size_t o = ((size_t)b * DI + d) * NL + l;
    ew[o] = expf(dt * Aval);
    bx[o] = dt * Bv * xs[(size_t)d * NL + l];
  }
}

// ---------------------------------------------------------------------------
// 9) tree scan: up-sweep (atomic scatter-add) + down-sweep, 1 block/batch
// ---------------------------------------------------------------------------
__global__ void tree_scan_kernel(const int* __restrict__ parent,
                                 const int* __restrict__ order,
                                 const int* __restrict__ lvls,
                                 const int* __restrict__ mdep,
                                 const float* __restrict__ ew,
                                 const float* __restrict__ bx,
                                 float* __restrict__ up, float* __restrict__ ag) {
  int b = blockIdx.x;
  const int n = NL;
  const int* PAR = parent + (size_t)b * n;
  const int* ORD = order  + (size_t)b * n;
  const int* LS  = lvls   + (size_t)b * 4100;
  int md = mdep[b];
  const float* W = ew + (size_t)b * DI * n;
  const float* X = bx + (size_t)b * DI * n;
  float* U = up + (size_t)b * DI * n;
  float* A = ag + (size_t)b * DI * n;
  for (int i = threadIdx.x; i < DI * n; i += blockDim.x) U[i] = X[i];
  __syncthreads();
  for (int d = md; d >= 1; --d) {
    int s = LS[d], e = LS[d + 1], cnt = e - s;
    for (int t = threadIdx.x; t < cnt * DI; t += blockDim.x) {
      int c = t / cnt, i = t % cnt;
      int v = ORD[s + i], p = PAR[v];
      float w = W[(size_t)c * n + v], uv = U[(size_t)c * n + v];
      atomicAdd(&U[(size_t)c * n + p], w * uv);
    }
    __syncthreads();
  }
  for (int i = threadIdx.x; i < DI * n; i += blockDim.x) A[i] = U[i];
  __syncthreads();
  for (int d = 1; d <= md; ++d) {
    int s = LS[d], e = LS[d + 1], cnt = e - s;
    for (int t = threadIdx.x; t < cnt * DI; t += blockDim.x) {
      int c = t / cnt, i = t % cnt;
      int v = ORD[s + i], p = PAR[v];
      float w = W[(size_t)c * n + v], uv = U[(size_t)c * n + v];
      A[(size_t)c * n + v] = uv + w * (A[(size_t)c * n + p] - w * uv);
    }
    __syncthreads();
  }
}

// ---------------------------------------------------------------------------
// 10) fused: LN(h)*C + D*xs -> LN -> *z ; one block (192 threads) per (b,l)
// ---------------------------------------------------------------------------
__global__ void fuse_norm_kernel(const float* __restrict__ ag,
                                 const float* __restrict__ xc,
                                 const float* __restrict__ zs,
                                 const float* __restrict__ cs,
                                 const float* __restrict__ Ds,
                                 const float* __restrict__ g1, const float* __restrict__ b1,
                                 const float* __restrict__ g2, const float* __restrict__ b2,
                                 float* __restrict__ ybuf) {
  int bl = blockIdx.x;
  int b = bl / NL, l = bl % NL;
  int c = threadIdx.x;
  __shared__ float sh[DI];
  __shared__ float s_mu, s_rs;
  float h = ag[((size_t)b * DI + c) * NL + l];
  sh[c] = h; __syncthreads();
  if (c == 0) { float s = 0.f; for (int i = 0; i < DI; ++i) s += sh[i]; s_mu = s / DI; }
  __syncthreads();
  float d0 = h - s_mu;
  sh[c] = d0 * d0; __syncthreads();
  if (c == 0) { float s = 0.f; for (int i = 0; i < DI; ++i) s += sh[i]; s_rs = rsqrtf(s / DI + 1e-5f); }
  __syncthreads();
  float hn = d0 * s_rs * g1[c] + b1[c];
  float y1 = hn * cs[bl] + Ds[c] * xc[((size_t)b * DI + c) * NL + l];
  sh[c] = y1; __syncthreads();
  if (c == 0) { float s = 0.f; for (int i = 0; i < DI; ++i) s += sh[i]; s_mu = s / DI; }
  __syncthreads();
  d0 = y1 - s_mu;
  sh[c] = d0 * d0; __syncthreads();
  if (c == 0) { float s = 0.f; for (int i = 0; i < DI; ++i) s += sh[i]; s_rs = rsqrtf(s / DI + 1e-5f); }
  __syncthreads();
  float y2 = d0 * s_rs * g2[c] + b2[c];
  ybuf[(size_t)bl * DI + c] = y2 * zs[(size_t)bl * DI + c];
}

// ---------------------------------------------------------------------------
// 11) out_proj GEMM: Y[32768,192] @ W^T (W=(96,192)) -> out[32768,96], WMMA
// ---------------------------------------------------------------------------
__global__ void out_proj_gemm_kernel(const float* __restrict__ Y,
                                     const float* __restrict__ Wo,  // (96,192)
                                     float* __restrict__ out) {
  const int NT = DM / 16;                             // 6 N-tiles
  int wave = blockIdx.x * (blockDim.x >> 5) + (threadIdx.x >> 5);
  int lane = threadIdx.x & 31;
  int tm = (wave / NT) * 16;
  int tn = (wave % NT) * 16;
  int row  = lane & 15;
  int half = lane >> 4;
  v8f acc = {0.f,0.f,0.f,0.f,0.f,0.f,0.f,0.f};
  const float* arow = Y  + (size_t)(tm + row) * DI;
  const float* brow = Wo + (size_t)(tn + row) * DI;
  for (int k = 0; k < DI; k += 4) {
    int ka = k + half * 2;
    v2f a;  a.x  = arow[ka]; a.y  = arow[ka + 1];
    v2f bv; bv.x = brow[ka]; bv.y = brow[ka + 1];
    acc = __builtin_amdgcn_wmma_f32_16x16x4_f32(false, a, false, bv,
                                                (short)0, acc, false, false);
  }
  for (int g = 0; g < 8; ++g) {
    int r = tm + g + half * 8;
    out[(size_t)r * DM + tn + row] = acc[g];
  }
}

// ---------------------------------------------------------------------------
extern "C" void kernel_launch(void* const* d_in, const int* in_sizes, int n_in,
                              void* d_out, int out_size, void* d_ws, size_t ws_size,
                              hipStream_t stream) {
  (void)in_sizes; (void)n_in; (void)out_size; (void)ws_size;
  const float* x        = (const float*)d_in[0];
  const float* in_w     = (const float*)d_in[1];
  const float* conv_w   = (const float*)d_in[2];
  const float* conv_b   = (const float*)d_in[3];
  const float* xproj_w  = (const float*)d_in[4];
  const float* dt_w     = (const float*)d_in[5];
  const float* dt_b     = (const float*)d_in[6];
  const float* a_logs   = (const float*)d_in[7];
  const float* ds_par   = (const float*)d_in[8];
  const float* hn_g     = (const float*)d_in[9];
  const float* hn_b     = (const float*)d_in[10];
  const float* on_g     = (const float*)d_in[11];
  const float* on_b     = (const float*)d_in[12];
  const float* out_w    = (const float*)d_in[13];
  float* out = (float*)d_out;

  char* base = (char*)d_ws;
  size_t off = 0;
  auto alloc = [&](size_t bytes) -> void* {
    void* r = base + off;
    off = (off + bytes + 255) & ~(size_t)255;
    return r;
  };
  float*  xz    = (float*) alloc(sizeof(float)  * (size_t)NTOK * 2 * DI);
  float*  xc    = (float*) alloc(sizeof(float)  * (size_t)BSZ * DI * NL);
  float*  zs    = (float*) alloc(sizeof(float)  * (size_t)NTOK * DI);
  float*  ew    = (float*) alloc(sizeof(float)  * (size_t)BSZ * DI * NL);
  float*  bx    = (float*) alloc(sizeof(float)  * (size_t)BSZ * DI * NL);
  float*  up    = (float*) alloc(sizeof(float)  * (size_t)BSZ * DI * NL);
  float*  ag    = (float*) alloc(sizeof(float)  * (size_t)BSZ * DI * NL);
  float*  ybuf  = (float*) alloc(sizeof(float)  * (size_t)NTOK * DI);
  float*  csbuf = (float*) alloc(sizeof(float)  * (size_t)NTOK);
  double* invn  = (double*)alloc(sizeof(double) * (size_t)BSZ * NL);
  double* keys  = (double*)alloc(sizeof(double) * (size_t)BSZ * NEP);
  int*    eidx  = (int*)   alloc(sizeof(int)    * (size_t)BSZ * NEP);
  int*    par   = (int*)   alloc(sizeof(int)    * (size_t)BSZ * NL);
  int*    dep   = (int*)   alloc(sizeof(int)    * (size_t)BSZ * NL);
  int*    ord   = (int*)   alloc(sizeof(int)    * (size_t)BSZ * NL);
  int*    lvls  = (int*)   alloc(sizeof(int)    * (size_t)BSZ * 4100);
  int*    uf    = (int*)   alloc(sizeof(int)    * (size_t)BSZ * NL);
  int*    adjh  = (int*)   alloc(sizeof(int)    * (size_t)BSZ * NL);
  int*    adjn  = (int*)   alloc(sizeof(int)    * (size_t)BSZ * NEP);
  int*    adjt  = (int*)   alloc(sizeof(int)    * (size_t)BSZ * NEP);
  int*    mdep  = (int*)   alloc(sizeof(int)    * (size_t)BSZ);

  // 1) in_proj GEMM (WMMA f32): 2048 M-tiles * 24 N-tiles = 49152 waves
  in_proj_gemm_kernel<<<49152 / 8, 256, 0, stream>>>(x, in_w, xz);
  // 2) z = silu
  z_silu_kernel<<<(NTOK * DI + 255) / 256, 256, 0, stream>>>(xz, zs);
  // 3) depthwise conv + silu
  dwconv_silu_kernel<<<(BSZ * DI * NL + 255) / 256, 256, 0, stream>>>(xz, conv_w, conv_b, xc);
  // 4-7) tree build
  node_norm_kernel<<<(BSZ * NL + 255) / 256, 256, 0, stream>>>(xc, invn);
  edge_weight_kernel<<<(BSZ * NEP + 255) / 256, 256, 0, stream>>>(xc, invn, keys, eidx);
  sort_edges_kernel<<<BSZ, 1024, 0, stream>>>(keys, eidx);
  build_tree_kernel<<<1, BSZ, 0, stream>>>(eidx, par, dep, ord, lvls, uf, adjh, adjn, adjt, mdep);
  // 8) SSM prologue
  ssm_pre_kernel<<<(NTOK + 255) / 256, 256, 0, stream>>>(xc, xproj_w, dt_w, dt_b, a_logs, ew, bx, csbuf);
  // 9) tree scan
  tree_scan_kernel<<<BSZ, 1024, 0, stream>>>(par, ord, lvls, mdep, ew, bx, up, ag);
  // 10) fused norms/elementwise
  fuse_norm_kernel<<<NTOK, DI, 0, stream>>>(ag, xc, zs, csbuf, ds_par, hn_g, hn_b, on_g, on_b, ybuf);
  // 11) out_proj GEMM (WMMA f32): 2048 * 6 = 12288 waves
  out_proj_gemm_kernel<<<12288 / 8, 256, 0, stream>>>(ybuf, out_w, out);
}